// DifferentiableRFAGNN_78769700208725
// MI455X (gfx1250) — compile-verified
//
#include <hip/hip_runtime.h>
#include <math.h>

#define N_NODES 2048
#define BATCH   8
#define HDIM    64
#define ALPHA   0.85f
#define CAP     128      // per-row edge capacity (deg ~ Binom(2048,1/64), mean 32)
#define ITERS   128      // 0.85^128 ~ 9e-10 : below fp32 resolution (must be even)

typedef __attribute__((ext_vector_type(2))) float v2f;
typedef __attribute__((ext_vector_type(8))) float v8f;

// ---------------------------------------------------------------------------
// 1. Compact signed adjacency into deterministic ELL (wave per row, ballot scan)
// ---------------------------------------------------------------------------
__global__ __launch_bounds__(256)
void build_edges_kernel(const float* __restrict__ adj,
                        int* __restrict__ cnt,
                        int* __restrict__ col,
                        float* __restrict__ eval_)
{
    const int lane = threadIdx.x & 31;
    const int row  = blockIdx.x * 8 + (threadIdx.x >> 5);
    const float* arow = adj + (size_t)row * N_NODES;
    int base = 0;
    for (int c0 = 0; c0 < N_NODES; c0 += 32) {
        __builtin_prefetch(arow + c0 + 256, 0, 0);   // global_prefetch_b8
        float v  = arow[c0 + lane];
        bool  nz = fabsf(v) > 1e-6f;
        unsigned mask = (unsigned)__ballot(nz);      // wave32: low 32 bits
        int pre = __popc(mask & ((1u << lane) - 1u));
        int slot = base + pre;
        if (nz && slot < CAP) {
            col  [(size_t)row * CAP + slot] = c0 + lane;
            eval_[(size_t)row * CAP + slot] = v;
        }
        base += __popc(mask);
    }
    if (lane == 0) cnt[row] = (base < CAP) ? base : CAP;
}

// ---------------------------------------------------------------------------
// 2. Encoder: h[b,n,j] = relu(x*W_enc[0,j] + t*W_enc[1,j] + b_enc[j])
// ---------------------------------------------------------------------------
__global__ __launch_bounds__(256)
void encoder_kernel(const float* __restrict__ x, const float* __restrict__ t,
                    const float* __restrict__ W_enc, const float* __restrict__ b_enc,
                    float* __restrict__ h)
{
    int idx = blockIdx.x * blockDim.x + threadIdx.x;   // over B*N*H
    int bn = idx >> 6, j = idx & 63;
    float v = x[bn] * W_enc[j] + t[bn] * W_enc[64 + j] + b_enc[j];
    h[idx] = fmaxf(v, 0.0f);
}

// ---------------------------------------------------------------------------
// 3. Sparse aggregation: agg[b,i,:] = sum_{j in nbr(i)} h[b,j,:]
//    wave per (b,i); each lane owns a float2 slice of H
// ---------------------------------------------------------------------------
__global__ __launch_bounds__(256)
void spmm_kernel(const float* __restrict__ h_in,
                 const int* __restrict__ cnt, const int* __restrict__ col,
                 float* __restrict__ h_out)
{
    const int wid  = blockIdx.x * 8 + (threadIdx.x >> 5);   // 0..B*N-1
    const int lane = threadIdx.x & 31;
    const int b = wid / N_NODES, i = wid % N_NODES;
    const int c = cnt[i];
    const int* cols = col + (size_t)i * CAP;
    const float* hb = h_in + (size_t)b * N_NODES * HDIM;
    float2 acc = {0.0f, 0.0f};
    for (int e = 0; e < c; ++e) {
        const float2* hj = (const float2*)(hb + (size_t)cols[e] * HDIM);
        float2 v = hj[lane];
        acc.x += v.x; acc.y += v.y;
    }
    ((float2*)(h_out + (size_t)wid * HDIM))[lane] = acc;
}

// ---------------------------------------------------------------------------
// 4. Dense (R,64)@(64,64) GEMM on fp32 WMMA (V_WMMA_F32_16X16X4_F32).
//    Block = 4 waves; wave w computes the 16x16 tile at cols [16w,16w+16).
//    A frag (16x4): lanes 0-15 hold K=k0,k0+1 ; lanes 16-31 hold K=k0+2,k0+3.
//    C/D: VGPR r -> row r + 8*(lane>=16), col = lane&15   (ISA 7.12.2)
// ---------------------------------------------------------------------------
__global__ __launch_bounds__(128)
void gemm64_wmma_kernel(const float* __restrict__ X,   // (Rtot,64)
                        const float* __restrict__ W,   // (64,64)
                        const float* __restrict__ bias,// (64) or nullptr
                        float* __restrict__ Y,         // (Rtot,64)
                        int relu)
{
    const int wave = threadIdx.x >> 5;
    const int lane = threadIdx.x & 31;
    const int m    = lane & 15;
    const int hi   = lane >> 4;
    const int row0 = blockIdx.x * 16;
    const int col  = wave * 16 + m;
    const float* Xrow = X + (size_t)(row0 + m) * HDIM;

    v8f acc = {};
#pragma unroll
    for (int kk = 0; kk < 16; ++kk) {
        const int k0 = kk * 4 + 2 * hi;
        v2f a, bf;
        a.x  = Xrow[k0];
        a.y  = Xrow[k0 + 1];
        bf.x = W[(size_t)k0 * HDIM + col];
        bf.y = W[(size_t)(k0 + 1) * HDIM + col];
        acc = __builtin_amdgcn_wmma_f32_16x16x4_f32(
                  /*neg_a=*/false, a, /*neg_b=*/false, bf,
                  /*c_mod=*/(short)0, acc, /*reuse_a=*/false, /*reuse_b=*/false);
    }
    const float bv = bias ? bias[col] : 0.0f;
#pragma unroll
    for (int r = 0; r < 8; ++r) {
        float v = acc[r] + bv;
        if (relu) v = fmaxf(v, 0.0f);
        Y[(size_t)(row0 + r + 8 * hi) * HDIM + col] = v;
    }
}

// ---------------------------------------------------------------------------
// 5. Edge gating + row normalization, packed output:
//    a    = adj_ij * sigmoid(<h_src[i], h_dst[j]> + gate_b)
//    S_ij = alpha * a / max(sum_j |a|, 1e-9)
//    epk[b,i,e] = (col_bits, S_ij)  as float2    (wave per (b,i))
// ---------------------------------------------------------------------------
__global__ __launch_bounds__(256)
void gate_kernel(const float* __restrict__ hsrc, const float* __restrict__ hdst,
                 const int* __restrict__ cnt, const int* __restrict__ col,
                 const float* __restrict__ eval_, const float* __restrict__ gate_b,
                 float2* __restrict__ epk)
{
    const int wid  = blockIdx.x * 8 + (threadIdx.x >> 5);
    const int lane = threadIdx.x & 31;
    const int b = wid / N_NODES, i = wid % N_NODES;
    const int c = cnt[i];
    const float gb = gate_b[0];
    const float* hs = hsrc + (size_t)(b * N_NODES + i) * HDIM;
    const float* hdb = hdst + (size_t)b * N_NODES * HDIM;
    const int*   cols = col  + (size_t)i * CAP;
    const float* ev   = eval_ + (size_t)i * CAP;

    float a[CAP / 32];
    int   jc[CAP / 32];
    float asum = 0.0f;
#pragma unroll
    for (int q = 0; q < CAP / 32; ++q) {
        const int e = q * 32 + lane;
        float av = 0.0f;
        int   jv = 0;
        if (e < c) {
            jv = cols[e];
            const float* hd = hdb + (size_t)jv * HDIM;
            float dot = 0.0f;
#pragma unroll 8
            for (int k = 0; k < HDIM; ++k) dot += hs[k] * hd[k];
            const float g = 1.0f / (1.0f + __expf(-(dot + gb)));
            av = ev[e] * g;
        }
        a[q] = av;
        jc[q] = jv;
        asum += fabsf(av);
    }
#pragma unroll
    for (int off = 16; off > 0; off >>= 1)
        asum += __shfl_xor(asum, off, 32);
    const float inv = ALPHA / fmaxf(asum, 1e-9f);
    float2* sv = epk + (size_t)wid * CAP;
#pragma unroll
    for (int q = 0; q < CAP / 32; ++q) {
        float2 pr;
        pr.x = __int_as_float(jc[q]);
        pr.y = a[q] * inv;
        sv[q * 32 + lane] = pr;
    }
}

// ---------------------------------------------------------------------------
// 6. Persistent Neumann solve, one workgroup per batch, y lives in LDS.
//    y <- t + S y  repeated ITERS times (LDS ping-pong, s_barrier between
//    sweeps); epilogue writes out = x + y directly.
// ---------------------------------------------------------------------------
__global__ __launch_bounds__(1024)
void neumann_persistent_kernel(const float* __restrict__ x,
                               const float* __restrict__ t,
                               const int* __restrict__ cnt,
                               const float2* __restrict__ epk,
                               float* __restrict__ out)
{
    __shared__ float ybufA[N_NODES];   // 8 KB
    __shared__ float ybufB[N_NODES];   // 8 KB
    const int b   = blockIdx.x;
    const int tid = threadIdx.x;
    const int i0 = tid, i1 = tid + 1024;             // 2 rows per thread

    // per-row constants cached in registers
    const int c0 = cnt[i0], c1 = cnt[i1];
    const float2* sv0 = epk + ((size_t)b * N_NODES + i0) * CAP;
    const float2* sv1 = epk + ((size_t)b * N_NODES + i1) * CAP;
    const float t0 = t[(size_t)b * N_NODES + i0];
    const float t1 = t[(size_t)b * N_NODES + i1];

    ybufA[i0] = t0;
    ybufA[i1] = t1;
    __syncthreads();

    for (int it = 0; it < ITERS; ++it) {
        const float* yin  = (it & 1) ? ybufB : ybufA;
        float*       yout = (it & 1) ? ybufA : ybufB;
        float acc0 = t0, acc1 = t1;
        for (int e = 0; e < c0; ++e) {
            float2 p = sv0[e];
            acc0 += p.y * yin[__float_as_int(p.x)];  // ds gather
        }
        for (int e = 0; e < c1; ++e) {
            float2 p = sv1[e];
            acc1 += p.y * yin[__float_as_int(p.x)];
        }
        yout[i0] = acc0;
        yout[i1] = acc1;
        __syncthreads();
    }
    // ITERS even -> final vector in ybufA
    out[(size_t)b * N_NODES + i0] = x[(size_t)b * N_NODES + i0] + ybufA[i0];
    out[(size_t)b * N_NODES + i1] = x[(size_t)b * N_NODES + i1] + ybufA[i1];
}

// ---------------------------------------------------------------------------
extern "C" void kernel_launch(void* const* d_in, const int* in_sizes, int n_in,
                              void* d_out, int out_size, void* d_ws, size_t ws_size,
                              hipStream_t stream)
{
    (void)in_sizes; (void)n_in; (void)out_size; (void)ws_size;
    const float* x_ctl   = (const float*)d_in[0];
    const float* targets = (const float*)d_in[1];
    const float* adj     = (const float*)d_in[2];
    const float* W_enc   = (const float*)d_in[3];
    const float* b_enc   = (const float*)d_in[4];
    const float* W_g1    = (const float*)d_in[5];
    const float* b_g1    = (const float*)d_in[6];
    const float* W_g2    = (const float*)d_in[7];
    const float* b_g2    = (const float*)d_in[8];
    const float* Ws      = (const float*)d_in[9];
    const float* Wd      = (const float*)d_in[10];
    const float* gate_b  = (const float*)d_in[11];

    // workspace carve-out (256B aligned slabs)
    char* p = (char*)d_ws;
    size_t off = 0;
    auto take = [&](size_t bytes) {
        char* r = p + off;
        off += (bytes + 255) & ~(size_t)255;
        return r;
    };
    int*    cnt  = (int*)   take((size_t)N_NODES * 4);
    int*    col  = (int*)   take((size_t)N_NODES * CAP * 4);
    float*  ev   = (float*) take((size_t)N_NODES * CAP * 4);
    float*  h0   = (float*) take((size_t)BATCH * N_NODES * HDIM * 4);
    float*  h1   = (float*) take((size_t)BATCH * N_NODES * HDIM * 4);
    float*  hsrc = (float*) take((size_t)BATCH * N_NODES * HDIM * 4);
    float*  hdst = (float*) take((size_t)BATCH * N_NODES * HDIM * 4);
    float2* epk  = (float2*)take((size_t)BATCH * N_NODES * CAP * 8);

    const int BN = BATCH * N_NODES;           // 16384

    // 1. edge extraction (deterministic, order-preserving)
    build_edges_kernel<<<N_NODES / 8, 256, 0, stream>>>(adj, cnt, col, ev);

    // 2. encoder
    encoder_kernel<<<(BN * HDIM) / 256, 256, 0, stream>>>(x_ctl, targets, W_enc, b_enc, h0);

    // 3-4. two rounds of aggregate + WMMA GEMM (+bias, relu)
    spmm_kernel<<<BN / 8, 256, 0, stream>>>(h0, cnt, col, h1);
    gemm64_wmma_kernel<<<BN / 16, 128, 0, stream>>>(h1, W_g1, b_g1, h0, 1);
    spmm_kernel<<<BN / 8, 256, 0, stream>>>(h0, cnt, col, h1);
    gemm64_wmma_kernel<<<BN / 16, 128, 0, stream>>>(h1, W_g2, b_g2, h0, 1);

    // 5. src/dst projections (WMMA, no bias/relu)
    gemm64_wmma_kernel<<<BN / 16, 128, 0, stream>>>(h0, Ws, nullptr, hsrc, 0);
    gemm64_wmma_kernel<<<BN / 16, 128, 0, stream>>>(h0, Wd, nullptr, hdst, 0);

    // 6. edge gating + alpha/out_deg row scaling -> packed (col, S) pairs
    gate_kernel<<<BN / 8, 256, 0, stream>>>(hsrc, hdst, cnt, col, ev, gate_b, epk);

    // 7. persistent fixed-point solve in LDS + fused output add (1 launch)
    neumann_persistent_kernel<<<BATCH, 1024, 0, stream>>>(x_ctl, targets, cnt, epk, (float*)d_out);
}